// SimpleHeteroGNN_72438918414564
// MI455X (gfx1250) — compile-verified
//
#include <hip/hip_runtime.h>
#include <hip/hip_bf16.h>

typedef float v2f __attribute__((ext_vector_type(2)));
typedef float v8f __attribute__((ext_vector_type(8)));

__device__ __forceinline__ v8f wmma4_f32(v2f a, v2f b, v8f c) {
    // D = A(16x4,f32) * B(4x16,f32) + C(16x16,f32)
    return __builtin_amdgcn_wmma_f32_16x16x4_f32(
        /*neg_a=*/false, a, /*neg_b=*/false, b,
        /*c_mod=*/(short)0, c, /*reuse_a=*/false, /*reuse_b=*/false);
}

#define H 64

// ---------------------------------------------------------------------------
// Edge aggregation: acc[dst] += relu(emb[ids[src]]) ; deg[dst] += 1
// One wave (32 lanes) per edge; each lane handles 2 channels via float2.
// Index loads are wave-uniform (1 cacheline request each); the 256B embedding
// row is a single coalesced b64-per-lane gather that stays L2-resident.
// ---------------------------------------------------------------------------
__global__ void __launch_bounds__(256)
gnn_edge_agg(const int* __restrict__ src, const int* __restrict__ dst,
             const int* __restrict__ ids, const float* __restrict__ emb,
             float* __restrict__ acc, float* __restrict__ deg, int nE)
{
    const int wave = blockIdx.x * 8 + (threadIdx.x >> 5);   // one edge per wave
    const int lane = threadIdx.x & 31;
    if (wave >= nE) return;

    const int s = src[wave];
    const int d = dst[wave];
    const int r = ids[s];

    const v2f* row = (const v2f*)(emb + (size_t)r * H);
    v2f v = row[lane];
    float* a0 = acc + (size_t)d * H + 2 * lane;
    atomicAdd(a0,     fmaxf(v.x, 0.0f));
    atomicAdd(a0 + 1, fmaxf(v.y, 0.0f));
    if (lane == 0) atomicAdd(&deg[d], 1.0f);
}

// ---------------------------------------------------------------------------
// Fused per-diag pipeline, one wave per 16 diag rows:
//   Y  = 0.5*(Sb@Wb1 + degb (x) bb1 + St@Wt1 + degt (x) bt1)     [16x64]
//   Z  = relu(Y@Wu1 + bu1)                                        [16x64]
//   Hd = relu(Z@oW1 + ob1)                                        [16x32]
//   outval = sigmoid(Hd . oW2 + ob2)                              [16]
// All matmuls via V_WMMA_F32_16X16X4_F32 (K=64 -> 16 k-steps).
// ---------------------------------------------------------------------------
__global__ void __launch_bounds__(128)
gnn_diag_head(const float* __restrict__ Sb, const float* __restrict__ St,
              const float* __restrict__ degb, const float* __restrict__ degt,
              const float* __restrict__ Wb1, const float* __restrict__ bb1,
              const float* __restrict__ Wt1, const float* __restrict__ bt1,
              const float* __restrict__ Wu1, const float* __restrict__ bu1,
              const float* __restrict__ oW1, const float* __restrict__ ob1,
              const float* __restrict__ oW2, const float* __restrict__ ob2,
              float* __restrict__ outval, int nd, int ngroups)
{
    __shared__ float Ybuf[4][16 * H];
    __shared__ float Zbuf[4][16 * H];
    __shared__ float Hbuf[4][16 * 32];

    const int wid  = threadIdx.x >> 5;
    const int lane = threadIdx.x & 31;
    const int g    = blockIdx.x * 4 + wid;
    if (g >= ngroups) return;                 // wave-uniform: EXEC stays full for WMMA

    const int m0 = g * 16;
    const int M  = lane & 15;                 // A-matrix row / B,C column
    const int h  = lane >> 4;                 // half-wave selector
    float* Y  = Ybuf[wid];
    float* Z  = Zbuf[wid];
    float* Hd = Hbuf[wid];

    // ---- Stage 1: Y = 0.5*(Sb@Wb1 + St@Wt1 + deg-weighted biases) ----
    for (int n = 0; n < 4; ++n) {
        const int n0 = n * 16;
        v8f c = {};
        for (int kk = 0; kk < 16; ++kk) {
            const int kb = kk * 4 + 2 * h;    // this lane-half holds K = kb, kb+1
            v2f a = *(const v2f*)(Sb + (m0 + M) * H + kb);
            v2f b;
            b.x = Wb1[kb * H + n0 + M];
            b.y = Wb1[(kb + 1) * H + n0 + M];
            c = wmma4_f32(a, b, c);
            v2f a2 = *(const v2f*)(St + (m0 + M) * H + kb);
            v2f b2;
            b2.x = Wt1[kb * H + n0 + M];
            b2.y = Wt1[(kb + 1) * H + n0 + M];
            c = wmma4_f32(a2, b2, c);
        }
        for (int v = 0; v < 8; ++v) {
            const int Mr = v + 8 * h;         // C-matrix row
            float y = 0.5f * (c[v] + degb[m0 + Mr] * bb1[n0 + M]
                                   + degt[m0 + Mr] * bt1[n0 + M]);
            Y[Mr * H + n0 + M] = y;
        }
    }

    // ---- Stage 2: Z = relu(Y @ Wu1 + bu1) ----
    for (int n = 0; n < 4; ++n) {
        const int n0 = n * 16;
        v8f c = {};
        for (int kk = 0; kk < 16; ++kk) {
            const int kb = kk * 4 + 2 * h;
            v2f a;
            a.x = Y[M * H + kb];
            a.y = Y[M * H + kb + 1];
            v2f b;
            b.x = Wu1[kb * H + n0 + M];
            b.y = Wu1[(kb + 1) * H + n0 + M];
            c = wmma4_f32(a, b, c);
        }
        for (int v = 0; v < 8; ++v) {
            const int Mr = v + 8 * h;
            Z[Mr * H + n0 + M] = fmaxf(c[v] + bu1[n0 + M], 0.0f);
        }
    }

    // ---- Stage 3: Hd = relu(Z @ oW1 + ob1), oW1 is 64x32 ----
    for (int n = 0; n < 2; ++n) {
        const int n0 = n * 16;
        v8f c = {};
        for (int kk = 0; kk < 16; ++kk) {
            const int kb = kk * 4 + 2 * h;
            v2f a;
            a.x = Z[M * H + kb];
            a.y = Z[M * H + kb + 1];
            v2f b;
            b.x = oW1[kb * 32 + n0 + M];
            b.y = oW1[(kb + 1) * 32 + n0 + M];
            c = wmma4_f32(a, b, c);
        }
        for (int v = 0; v < 8; ++v) {
            const int Mr = v + 8 * h;
            Hd[Mr * 32 + n0 + M] = fmaxf(c[v] + ob1[n0 + M], 0.0f);
        }
    }

    // ---- Stage 4: outval = sigmoid(Hd . oW2 + ob2) ----
    if (lane < 16) {
        float acc = ob2[0];
        for (int j = 0; j < 32; ++j) acc += Hd[lane * 32 + j] * oW2[j];
        const int row = m0 + lane;
        if (row < nd) outval[row] = 1.0f / (1.0f + expf(-acc));
    }
}

// ---------------------------------------------------------------------------
// Final gather: out[i] = outval[inst2type[i]]
// ---------------------------------------------------------------------------
__global__ void __launch_bounds__(256)
gnn_inst_gather(const int* __restrict__ inst2type, const float* __restrict__ outval,
                float* __restrict__ out, int n)
{
    int i = blockIdx.x * blockDim.x + threadIdx.x;
    if (i < n) out[i] = outval[inst2type[i]];
}

extern "C" void kernel_launch(void* const* d_in, const int* in_sizes, int n_in,
                              void* d_out, int out_size, void* d_ws, size_t ws_size,
                              hipStream_t stream) {
    // Input order per setup_inputs():
    const int*   bene_ids      = (const int*)d_in[0];
    // d_in[1] = diag_ids (dead code: conv ignores diag input)
    const int*   treatment_ids = (const int*)d_in[2];
    const int*   b2d_src       = (const int*)d_in[3];
    const int*   b2d_dst       = (const int*)d_in[4];
    const int*   t2d_src       = (const int*)d_in[5];
    const int*   t2d_dst       = (const int*)d_in[6];
    const int*   inst2type     = (const int*)d_in[7];
    const float* bene_emb      = (const float*)d_in[8];
    // d_in[9] = diag_emb (dead)
    const float* treat_emb     = (const float*)d_in[10];
    // d_in[11..16] = layer-0 weights (dead: layer-0 diag output is overwritten)
    const float* Wb1 = (const float*)d_in[17];
    const float* bb1 = (const float*)d_in[18];
    const float* Wt1 = (const float*)d_in[19];
    const float* bt1 = (const float*)d_in[20];
    const float* Wu1 = (const float*)d_in[21];
    const float* bu1 = (const float*)d_in[22];
    const float* oW1 = (const float*)d_in[23];
    const float* ob1 = (const float*)d_in[24];
    const float* oW2 = (const float*)d_in[25];
    const float* ob2 = (const float*)d_in[26];
    float* out = (float*)d_out;

    const int Eb     = in_sizes[3];             // 1,000,000
    const int Et     = in_sizes[5];             // 1,000,000
    const int nd     = in_sizes[1];             // N_DIAG = 10,000
    const int n_inst = in_sizes[7];             // 500,000
    const int ndp    = (nd + 15) & ~15;         // pad to WMMA tile rows

    // Workspace layout (floats)
    float* Sb     = (float*)d_ws;               // [ndp*64] sum relu(bene rows) per diag
    float* St     = Sb + (size_t)ndp * H;       // [ndp*64]
    float* degb   = St + (size_t)ndp * H;       // [ndp]
    float* degt   = degb + ndp;                 // [ndp]
    float* outval = degt + ndp;                 // [ndp]
    size_t zero_bytes = ((size_t)2 * ndp * H + 2 * ndp) * sizeof(float);
    hipMemsetAsync(d_ws, 0, zero_bytes, stream);

    // Edge aggregation (the bandwidth/atomic-bound phase): 1 wave per edge,
    // 8 waves per 256-thread block.
    {
        int blocks_b = (Eb + 7) / 8;
        gnn_edge_agg<<<blocks_b, 256, 0, stream>>>(b2d_src, b2d_dst, bene_ids,
                                                   bene_emb, Sb, degb, Eb);
        int blocks_t = (Et + 7) / 8;
        gnn_edge_agg<<<blocks_t, 256, 0, stream>>>(t2d_src, t2d_dst, treatment_ids,
                                                   treat_emb, St, degt, Et);
    }

    // Fused WMMA diag pipeline: one wave per 16 diag rows, 4 waves per block
    {
        int ngroups = ndp / 16;                 // 625
        int blocks  = (ngroups + 3) / 4;
        gnn_diag_head<<<blocks, 128, 0, stream>>>(Sb, St, degb, degt,
                                                  Wb1, bb1, Wt1, bt1, Wu1, bu1,
                                                  oW1, ob1, oW2, ob2,
                                                  outval, nd, ngroups);
    }

    // Final scalar gather
    {
        int threads = 256;
        int blocks = (n_inst + threads - 1) / threads;
        gnn_inst_gather<<<blocks, threads, 0, stream>>>(inst2type, outval, out, n_inst);
    }
}